// LinearDynamicalSystem_65180423684626
// MI455X (gfx1250) — compile-verified
//
#include <hip/hip_runtime.h>

typedef __attribute__((ext_vector_type(2))) float v2f;
typedef __attribute__((ext_vector_type(8))) float v8f;

#define NT 8192
#define NB 2048
#define CH 16
#define NCHUNK (NT / CH)   // 512
#define NB4 (NB / 4)       // 512

// Compute chunk-local ARMA coefficient tables into LDS:
//   h[t]  : impulse response of 1/(1 + f1 z^-1 + f2 z^-2)
//   g[t]  = b0*h[t] + b1*h[t-1]   (zero-state Toeplitz column)
//   p1[t] : response to y[-1]=1,  p2[t] : response to y[-2]=1
//   q[t]  = b1*h[t]               (response to u[-1]=1)
// sgz is zero-padded: sgz[0..CH-1] = 0, sgz[CH+t] = g[t]  (branchless Toeplitz).
__device__ __forceinline__ void compute_coeffs(const float* b, const float* f,
                                               float* sgz, float* sp1,
                                               float* sp2, float* sq) {
  if (threadIdx.x == 0) {
    float f1 = f[0], f2 = f[1], b0 = b[0], b1 = b[1];
    float hm1 = 0.f, hm2 = 0.f;       // h[t-1], h[t-2]
    float p1m1 = 1.f, p1m2 = 0.f;     // p1[-1]=1, p1[-2]=0
    float p2m1 = 0.f, p2m2 = 1.f;     // p2[-1]=0, p2[-2]=1
    for (int t = 0; t < CH; ++t) {
      float ht  = (t == 0 ? 1.0f : 0.0f) - f1 * hm1 - f2 * hm2;
      float p1t = -f1 * p1m1 - f2 * p1m2;
      float p2t = -f1 * p2m1 - f2 * p2m2;
      sgz[t]      = 0.0f;             // zero pad (negative diagonals)
      sgz[CH + t] = b0 * ht + b1 * hm1;
      sq[t]  = b1 * ht;
      sp1[t] = p1t;
      sp2[t] = p2t;
      hm2 = hm1;  hm1 = ht;
      p1m2 = p1m1; p1m1 = p1t;
      p2m2 = p2m1; p2m1 = p2t;
    }
  }
  __syncthreads();
}

// ---------------------------------------------------------------------------
// Phase A: zero-state response per 16x16 tile:  Z = G @ U
// One wave per tile; 4 chained V_WMMA_F32_16X16X4_F32 (K=16 total).
// All A fetches (branchless LDS) and all 8 B global loads are hoisted ahead
// of the WMMA chain so loads clause together and latency overlaps.
// Grid: 512 time-tiles * 128 col-tiles = 65536 waves = 8192 blocks of 256.
// ---------------------------------------------------------------------------
__global__ void __launch_bounds__(256)
arma_phaseA_wmma(const float* __restrict__ u, const float* __restrict__ b,
                 const float* __restrict__ f, float* __restrict__ z) {
  __shared__ float sgz[2 * CH], sp1[CH], sp2[CH], sq[CH];
  compute_coeffs(b, f, sgz, sp1, sp2, sq);

  int wave   = (int)((blockIdx.x * blockDim.x + threadIdx.x) >> 5);
  int lane   = (int)(threadIdx.x & 31);
  int tile_n = wave & (NB / 16 - 1);   // 0..127
  int tile_t = wave >> 7;              // 0..511
  int m  = lane & 15;                  // row (A) / col (B,C)
  int h2 = lane >> 4;                  // lane half
  int col = tile_n * 16 + m;
  int t0  = tile_t * CH;

  // A operands: G[m][k] = g[m-k], zero-padded table -> branchless
  v2f a[4];
#pragma unroll
  for (int s = 0; s < 4; ++s) {
    int k0 = 4 * s + 2 * h2;           // global K index of this lane's pair
    a[s].x = sgz[CH + m - k0];
    a[s].y = sgz[CH + m - k0 - 1];
  }

  // B operands: U[t0+k0 : t0+k0+2, col] — 8 global loads, hoisted
  v2f bv[4];
#pragma unroll
  for (int s = 0; s < 4; ++s) {
    int k0 = 4 * s + 2 * h2;
    const float* p = u + (size_t)(t0 + k0) * NB + col;
    bv[s].x = p[0];
    bv[s].y = p[NB];
  }

  v8f c = {};
#pragma unroll
  for (int s = 0; s < 4; ++s) {
    c = __builtin_amdgcn_wmma_f32_16x16x4_f32(
        /*neg_a=*/false, a[s], /*neg_b=*/false, bv[s],
        /*c_mod=*/(short)0, c, /*reuse_a=*/false, /*reuse_b=*/false);
  }

  // C layout: lanes 0-15 hold M=r (VGPR r), lanes 16-31 hold M=8+r
#pragma unroll
  for (int r = 0; r < 8; ++r) {
    z[(size_t)(t0 + 8 * h2 + r) * NB + col] = c[r];
  }
}

// ---------------------------------------------------------------------------
// Phase B: sequential chunk-boundary state scan (512 steps per column).
// State entering chunk c: (ym1, ym2, um1). Update uses Z rows 14,15 + u row 15.
// 2048 threads (8 blocks x 256), one column each.
// ---------------------------------------------------------------------------
__global__ void __launch_bounds__(256)
arma_phaseB_scan(const float* __restrict__ u, const float* __restrict__ z,
                 const float* __restrict__ b, const float* __restrict__ f,
                 const float* __restrict__ y_init, const float* __restrict__ u_init,
                 float* __restrict__ ws_ym1, float* __restrict__ ws_ym2,
                 float* __restrict__ ws_um1) {
  __shared__ float sgz[2 * CH], sp1[CH], sp2[CH], sq[CH];
  compute_coeffs(b, f, sgz, sp1, sp2, sq);

  int n = (int)(blockIdx.x * blockDim.x + threadIdx.x);  // 0..2047
  float ym1 = y_init[n * 2 + 0];
  float ym2 = y_init[n * 2 + 1];
  float um1 = u_init[n];

  const float A11 = sp1[15], A12 = sp2[15], Aq1 = sq[15];
  const float A21 = sp1[14], A22 = sp2[14], Aq2 = sq[14];

  for (int c = 0; c < NCHUNK; ++c) {
    ws_ym1[c * NB + n] = ym1;
    ws_ym2[c * NB + n] = ym2;
    ws_um1[c * NB + n] = um1;
    size_t base = (size_t)c * CH * NB;
    float z15 = z[base + 15 * NB + n];
    float z14 = z[base + 14 * NB + n];
    float u15 = u[base + 15 * NB + n];
    float ny1 = z15 + A11 * ym1 + A12 * ym2 + Aq1 * um1;  // y[t0+15]
    float ny2 = z14 + A21 * ym1 + A22 * ym2 + Aq2 * um1;  // y[t0+14]
    ym1 = ny1;
    ym2 = ny2;
    um1 = u15;
  }
}

// ---------------------------------------------------------------------------
// Phase C: fully parallel fix-up (in place over d_out):
//   y[t][n] = Z[t][n] + p1[t%16]*ym1[c][n] + p2[t%16]*ym2[c][n] + q[t%16]*um1[c][n]
// float4 along n: 8192*512 threads = 16384 blocks of 256.
// ---------------------------------------------------------------------------
__global__ void __launch_bounds__(256)
arma_phaseC_fixup(float* zy, const float* __restrict__ ws_ym1,
                  const float* __restrict__ ws_ym2, const float* __restrict__ ws_um1,
                  const float* __restrict__ b, const float* __restrict__ f) {
  __shared__ float sgz[2 * CH], sp1[CH], sp2[CH], sq[CH];
  compute_coeffs(b, f, sgz, sp1, sp2, sq);

  int idx = (int)(blockIdx.x * blockDim.x + threadIdx.x);  // 0 .. NT*NB4-1
  int n4 = idx & (NB4 - 1);
  int t  = idx >> 9;
  int tr = t & (CH - 1);
  int c  = t >> 4;
  int n  = n4 * 4;

  const float p1 = sp1[tr], p2 = sp2[tr], q = sq[tr];

  float4 zv = *(const float4*)(zy + (size_t)t * NB + n);
  float4 s1 = *(const float4*)(ws_ym1 + (size_t)c * NB + n);
  float4 s2 = *(const float4*)(ws_ym2 + (size_t)c * NB + n);
  float4 s3 = *(const float4*)(ws_um1 + (size_t)c * NB + n);

  float4 y;
  y.x = zv.x + p1 * s1.x + p2 * s2.x + q * s3.x;
  y.y = zv.y + p1 * s1.y + p2 * s2.y + q * s3.y;
  y.z = zv.z + p1 * s1.z + p2 * s2.z + q * s3.z;
  y.w = zv.w + p1 * s1.w + p2 * s2.w + q * s3.w;

  *(float4*)(zy + (size_t)t * NB + n) = y;
}

extern "C" void kernel_launch(void* const* d_in, const int* in_sizes, int n_in,
                              void* d_out, int out_size, void* d_ws, size_t ws_size,
                              hipStream_t stream) {
  const float* b_coeff = (const float*)d_in[0];   // (2,)
  const float* f_coeff = (const float*)d_in[1];   // (2,)
  const float* u_in    = (const float*)d_in[2];   // (8192, 2048)
  const float* y_init  = (const float*)d_in[3];   // (2048, 2)
  const float* u_init  = (const float*)d_in[4];   // (2048, 1)
  float* out = (float*)d_out;                     // (8192, 2048)

  // Workspace: three state arrays of NCHUNK*NB floats (4 MB each, 12 MB total)
  float* ws_ym1 = (float*)d_ws;
  float* ws_ym2 = ws_ym1 + (size_t)NCHUNK * NB;
  float* ws_um1 = ws_ym2 + (size_t)NCHUNK * NB;

  // Phase A: 65536 tile-waves -> 8192 blocks of 256 (8 waves each)
  arma_phaseA_wmma<<<dim3((NCHUNK * (NB / 16)) / 8), dim3(256), 0, stream>>>(
      u_in, b_coeff, f_coeff, out);

  // Phase B: one thread per column
  arma_phaseB_scan<<<dim3(NB / 256), dim3(256), 0, stream>>>(
      u_in, out, b_coeff, f_coeff, y_init, u_init, ws_ym1, ws_ym2, ws_um1);

  // Phase C: float4 per thread over the full output
  arma_phaseC_fixup<<<dim3((NT * NB4) / 256), dim3(256), 0, stream>>>(
      out, ws_ym1, ws_ym2, ws_um1, b_coeff, f_coeff);
}